// SelectiveLabelSmoothingLoss_16733192585810
// MI455X (gfx1250) — compile-verified
//
#include <hip/hip_runtime.h>
#include <math.h>

#define B_ROWS   8192
#define C_COLS   32000
#define A_SLOTS  64
#define EPS      0.1f
#define BLOCK    1024
#define NWAVES   (BLOCK / 32)

typedef __attribute__((ext_vector_type(2))) float v2f;
typedef __attribute__((ext_vector_type(8))) float v8f;

// gcc-style vector matching clang's V4i builtin parameter encoding
typedef int v4i __attribute__((vector_size(16)));
typedef __attribute__((address_space(1))) v4i as1_v4i;  // global
typedef __attribute__((address_space(3))) v4i as3_v4i;  // LDS

// ---------------------------------------------------------------------------
// Kernel 1: one workgroup per row.
//   1. Async-DMA the 125 KB row from global into LDS (gfx1250 async path,
//      tracked by ASYNCcnt).
//   2. Pass 1 (from LDS): row max.
//   3. Pass 2 (from LDS): sum of exp(x - max) accumulated with
//      V_WMMA_F32_16X16X4_F32 (A = exp tile, B = ones) -> lse.
//   4. Sparse tail: <=65 entries with O(A^2) dedupe, deterministic fold.
// ---------------------------------------------------------------------------
__global__ __launch_bounds__(BLOCK) void slsl_row_kernel(
    const float* __restrict__ pred,
    const int*   __restrict__ target,
    const int*   __restrict__ allowed,
    float*       __restrict__ row_loss)
{
    __shared__ float lds_row[C_COLS];      // 128000 B row buffer
    __shared__ float red[NWAVES];
    __shared__ float s_rowmax;
    __shared__ float s_lse;
    __shared__ float s_fill;
    __shared__ int   s_allowed[A_SLOTS];
    __shared__ int   s_target;
    __shared__ float s_contrib[A_SLOTS];

    const int row  = blockIdx.x;
    const int tid  = threadIdx.x;
    const int lane = tid & 31;
    const int wid  = tid >> 5;
    const float* prow = pred + (size_t)row * C_COLS;

    if (tid < A_SLOTS) s_allowed[tid] = allowed[(size_t)row * A_SLOTS + tid];
    if (tid == 0)      s_target = target[row];

    // ---- Stage the row into LDS (single HBM touch) ----
#if __has_builtin(__builtin_amdgcn_global_load_async_to_lds_b128)
    for (int i = tid; i < C_COLS / 4; i += BLOCK) {
        __builtin_amdgcn_global_load_async_to_lds_b128(
            (as1_v4i*)(prow + i * 4),
            (as3_v4i*)(&lds_row[i * 4]),
            0, 0);
    }
  #if __has_builtin(__builtin_amdgcn_s_wait_asynccnt)
    __builtin_amdgcn_s_wait_asynccnt(0);
  #else
    asm volatile("s_wait_asynccnt 0" ::: "memory");
  #endif
#else
    for (int i = tid; i < C_COLS / 4; i += BLOCK) {
        ((float4*)lds_row)[i] = ((const float4*)prow)[i];
    }
#endif
    __syncthreads();

    // ---- Pass 1: row max (from LDS) ----
    float m = -3.402823466e38f;
    for (int i = tid; i < C_COLS; i += BLOCK) m = fmaxf(m, lds_row[i]);
    for (int off = 16; off > 0; off >>= 1)
        m = fmaxf(m, __shfl_xor(m, off, 32));
    if (lane == 0) red[wid] = m;
    __syncthreads();
    if (tid == 0) {
        float rm = red[0];
        for (int w = 1; w < NWAVES; ++w) rm = fmaxf(rm, red[w]);
        s_rowmax = rm;
        int nv = 0;
        for (int i = 0; i < A_SLOTS; ++i) nv += (s_allowed[i] != -1) ? 1 : 0;
        s_fill = EPS / (float)nv;
    }
    __syncthreads();
    const float rm = s_rowmax;

    // ---- Pass 2: sum(exp(x - rm)) via V_WMMA_F32_16X16X4_F32 ----
    // Each WMMA: A(16x4 of exp values, 2 f32/lane) x B(4x16 ones) + C.
    // D[m][n] = per-row partial sums replicated over columns; folding all
    // 16 M rows of one column yields this wave's total.
    v8f acc = {};
    v2f ones; ones.x = 1.0f; ones.y = 1.0f;
    const int PER_ROUND = NWAVES * 64;              // 2048 elements / round
    const int ROUNDS = (C_COLS + PER_ROUND - 1) / PER_ROUND;  // 16
    for (int it = 0; it < ROUNDS; ++it) {
        int base = it * PER_ROUND + wid * 64 + lane * 2;
        v2f a;
        a.x = (base     < C_COLS) ? __expf(lds_row[base]     - rm) : 0.0f;
        a.y = (base + 1 < C_COLS) ? __expf(lds_row[base + 1] - rm) : 0.0f;
        acc = __builtin_amdgcn_wmma_f32_16x16x4_f32(
            false, a, false, ones, (short)0, acc, false, false);
    }
    float ssum = 0.0f;
    for (int k = 0; k < 8; ++k) ssum += acc[k];     // rows m(=0..7 / 8..15)
    ssum += __shfl_xor(ssum, 16, 32);               // + other half of M
    if (lane == 0) red[wid] = ssum;
    __syncthreads();
    if (tid == 0) {
        float tot = 0.0f;
        for (int w = 0; w < NWAVES; ++w) tot += red[w];   // fixed order
        s_lse = rm + __logf(tot);
    }
    __syncthreads();
    const float lse = s_lse;

    // ---- Sparse tail: distinct valid allowed classes (!= target) ----
    if (tid < A_SLOTS) {
        int c = s_allowed[tid];
        float contrib = 0.0f;
        if (c != -1 && c != s_target) {
            bool dup = false;
            for (int j = 0; j < tid; ++j) dup |= (s_allowed[j] == c);
            if (!dup) {
                float f = s_fill;
                contrib = f * __logf(f) - f * (lds_row[c] - lse);
            }
        }
        s_contrib[tid] = contrib;
    }
    __syncthreads();
    if (tid == 0) {
        const float ome = 1.0f - EPS;
        float total = ome * __logf(ome) - ome * (lds_row[s_target] - lse);
        for (int i = 0; i < A_SLOTS; ++i) total += s_contrib[i];
        row_loss[row] = total;
    }
}

// ---------------------------------------------------------------------------
// Kernel 2: deterministic fold of 8192 per-row losses -> scalar / B.
// ---------------------------------------------------------------------------
__global__ __launch_bounds__(256) void slsl_reduce_kernel(
    const float* __restrict__ row_loss, float* __restrict__ out)
{
    __shared__ float s[256];
    int tid = threadIdx.x;
    float acc = 0.0f;
    for (int r = tid; r < B_ROWS; r += 256) acc += row_loss[r];  // fixed order
    s[tid] = acc;
    __syncthreads();
    if (tid == 0) {
        float tot = 0.0f;
        for (int i = 0; i < 256; ++i) tot += s[i];               // fixed order
        out[0] = tot / (float)B_ROWS;
    }
}

extern "C" void kernel_launch(void* const* d_in, const int* in_sizes, int n_in,
                              void* d_out, int out_size, void* d_ws, size_t ws_size,
                              hipStream_t stream) {
    const float* pred    = (const float*)d_in[0];
    const int*   target  = (const int*)d_in[1];
    const int*   allowed = (const int*)d_in[2];
    float* row_loss = (float*)d_ws;                 // 8192 floats of scratch

    slsl_row_kernel<<<B_ROWS, BLOCK, 0, stream>>>(pred, target, allowed, row_loss);
    slsl_reduce_kernel<<<1, 256, 0, stream>>>(row_loss, (float*)d_out);
}